// GAT_72035191489125
// MI455X (gfx1250) — compile-verified
//
#include <hip/hip_runtime.h>
#include <hip/hip_bf16.h>
#include <math.h>

typedef __attribute__((ext_vector_type(2))) float v2f;
typedef __attribute__((ext_vector_type(8))) float v8f;

#define NEG_SLOPE 0.2f

// Order-preserving float -> uint encoding so we can use integer atomicMax
// (global_atomic_max_u32) for the segment max over edges.
__device__ __forceinline__ unsigned int enc_f32(float f) {
    unsigned int u = __float_as_uint(f);
    return (u & 0x80000000u) ? ~u : (u | 0x80000000u);
}
__device__ __forceinline__ float dec_f32(unsigned int u) {
    u = (u & 0x80000000u) ? (u & 0x7FFFFFFFu) : ~u;
    return __uint_as_float(u);
}

// ---------------------------------------------------------------------------
// WMMA f32 GEMM: Out[N,M] = A[N,K] @ B[K,M], row-major. K AND M are template
// parameters so the fully-unrolled K-loop uses ONLY immediate-offset loads:
//   A: global_load_b64 base_a, off offset:k*4         (contiguous K pair)
//   B: global_load_b32 base_b, off offset:(k+j)*M*4   (max 65KB < 24-bit imm)
// -> no per-iteration address arithmetic, 32 straight-line
//    V_WMMA_F32_16X16X4_F32 ops that the scheduler pipelines against loads.
//
// No predication in the hot loop: out-of-range rows/cols are CLAMPED for
// address safety. A garbage A-row only affects output rows we never store;
// a garbage B-col only affects output cols we never store (rows/cols are
// independent in D = A*B), so clamping is correctness-preserving.
//
// A-frag layout (ISA 7.12.2, 32-bit A 16x4): lane l holds row (l&15),
//   VGPR0 = K (l>>4)*2, VGPR1 = K (l>>4)*2+1  -> contiguous pair = b64 load.
// B-frag symmetric (cols striped across lanes). C/D: 8 VGPRs, lane l holds
//   col (l&15), VGPR v -> row v + (l>>4)*8.
// One wave per 16x16 tile; blockDim = 32 * colTiles; gridDim = rowTiles.
// ---------------------------------------------------------------------------
template <int K, int M>
__global__ void wmma_gemm_kernel(const float* __restrict__ A,
                                 const float* __restrict__ B,
                                 float* __restrict__ Out,
                                 int Nrows) {
    const int wave = threadIdx.x >> 5;
    const int lane = threadIdx.x & 31;
    const int n0   = blockIdx.x * 16;
    const int c0   = wave * 16;
    const int row  = n0 + (lane & 15);
    const int col  = c0 + (lane & 15);
    const int kb   = (lane >> 4) * 2;       // 0 or 2

    // clamp (not predicate) for address safety
    const int rowc = (row < Nrows) ? row : (Nrows - 1);
    const int colc = (col < M) ? col : (M - 1);

    const float* arow = A + (long long)rowc * K + kb;     // 8B aligned
    const float* bcol = B + (long long)kb * M + colc;

    v8f acc = {};
    #pragma unroll
    for (int k = 0; k < K; k += 4) {
        const v2f a = *(const v2f*)(arow + k);            // global_load_b64
        v2f b;
        b.x = bcol[k * M];                                // imm offsets
        b.y = bcol[(k + 1) * M];
        acc = __builtin_amdgcn_wmma_f32_16x16x4_f32(
            /*neg_a=*/false, a, /*neg_b=*/false, b,
            /*c_mod=*/(short)0, acc, /*reuse_a=*/false, /*reuse_b=*/false);
    }

    if (col < M) {
        const int rbase = n0 + (lane >> 4) * 8;
        #pragma unroll
        for (int v = 0; v < 8; ++v) {
            const int r = rbase + v;
            if (r < Nrows) Out[(long long)r * M + col] = acc[v];
        }
    }
}

// ---------------------------------------------------------------------------
// Per-(node, head): a_src[n,h] = <h[n,h,:], att_src[h,:]>, same for a_dst.
// Also initializes the segment-max (encoded) and denom accumulators.
// ---------------------------------------------------------------------------
__global__ void attn_score_kernel(const float* __restrict__ h,
                                  const float* __restrict__ att_src,
                                  const float* __restrict__ att_dst,
                                  float* __restrict__ a_src,
                                  float* __restrict__ a_dst,
                                  unsigned int* __restrict__ m_enc,
                                  float* __restrict__ denom,
                                  long long N, int H, int C) {
    const long long idx = blockIdx.x * (long long)blockDim.x + threadIdx.x;
    if (idx >= N * H) return;
    const long long n  = idx / H;
    const int       hd = (int)(idx % H);
    const float* hp = h + n * (long long)(H * C) + (long long)hd * C;
    const float* as = att_src + (long long)hd * C;
    const float* ad = att_dst + (long long)hd * C;
    float s = 0.0f, d = 0.0f;
    for (int c = 0; c < C; ++c) {
        const float v = hp[c];
        s += v * as[c];
        d += v * ad[c];
    }
    a_src[idx] = s;
    a_dst[idx] = d;
    m_enc[idx] = 0u;     // encodes "below any real float"
    denom[idx] = 0.0f;
}

// ---------------------------------------------------------------------------
// Edge pass A: per-destination max of leaky_relu(a_src[src]+a_dst[dst]).
// Edges e < E come from edge_index; e >= E are self-loops (node e-E).
// ---------------------------------------------------------------------------
__global__ void edge_max_kernel(const long long* __restrict__ src_idx,
                                const long long* __restrict__ dst_idx,
                                const float* __restrict__ a_src,
                                const float* __restrict__ a_dst,
                                unsigned int* __restrict__ m_enc,
                                long long E, long long EN, int H) {
    const long long e = blockIdx.x * (long long)blockDim.x + threadIdx.x;
    if (e >= EN) return;
    long long s, d;
    if (e < E) { s = src_idx[e]; d = dst_idx[e]; } else { s = d = e - E; }
    for (int hd = 0; hd < H; ++hd) {
        float al = a_src[s * H + hd] + a_dst[d * H + hd];
        al = (al >= 0.0f) ? al : NEG_SLOPE * al;
        atomicMax(&m_enc[d * H + hd], enc_f32(al));
    }
}

// Decode encoded max back to float, in place.
__global__ void decode_m_kernel(unsigned int* __restrict__ m, long long NH) {
    const long long i = blockIdx.x * (long long)blockDim.x + threadIdx.x;
    if (i >= NH) return;
    const float f = dec_f32(m[i]);
    ((float*)m)[i] = f;
}

// Edge pass B: denom[dst,h] += exp(alpha - m[dst,h]).
__global__ void edge_sum_kernel(const long long* __restrict__ src_idx,
                                const long long* __restrict__ dst_idx,
                                const float* __restrict__ a_src,
                                const float* __restrict__ a_dst,
                                const float* __restrict__ m,
                                float* __restrict__ denom,
                                long long E, long long EN, int H) {
    const long long e = blockIdx.x * (long long)blockDim.x + threadIdx.x;
    if (e >= EN) return;
    long long s, d;
    if (e < E) { s = src_idx[e]; d = dst_idx[e]; } else { s = d = e - E; }
    for (int hd = 0; hd < H; ++hd) {
        float al = a_src[s * H + hd] + a_dst[d * H + hd];
        al = (al >= 0.0f) ? al : NEG_SLOPE * al;
        atomicAdd(&denom[d * H + hd], __expf(al - m[d * H + hd]));
    }
}

// Reference adds bias AFTER aggregation: seed the accumulator with bias.
__global__ void init_bias_kernel(float* __restrict__ out,
                                 const float* __restrict__ bias,
                                 long long N, int HC) {
    const long long i = blockIdx.x * (long long)blockDim.x + threadIdx.x;
    if (i >= N * HC) return;
    out[i] = bias[i % HC];
}

// ---------------------------------------------------------------------------
// Edge pass C: one wave per edge. Lane covers channels ch, ch+32, ...
// coefficient = exp(alpha-m)/(denom+1e-16); out[dst,ch] += h[src,ch]*coef.
// ---------------------------------------------------------------------------
__global__ void edge_agg_kernel(const long long* __restrict__ src_idx,
                                const long long* __restrict__ dst_idx,
                                const float* __restrict__ a_src,
                                const float* __restrict__ a_dst,
                                const float* __restrict__ m,
                                const float* __restrict__ denom,
                                const float* __restrict__ h,
                                float* __restrict__ out,
                                long long E, long long EN, int H, int C) {
    const int lane = threadIdx.x & 31;
    const long long e = blockIdx.x * (long long)(blockDim.x >> 5)
                      + (long long)(threadIdx.x >> 5);
    if (e >= EN) return;
    long long s, d;
    if (e < E) { s = src_idx[e]; d = dst_idx[e]; } else { s = d = e - E; }
    const int HC = H * C;
    for (int ch = lane; ch < HC; ch += 32) {
        const int hd = (H == 1) ? 0 : (ch / C);
        float al = a_src[s * H + hd] + a_dst[d * H + hd];
        al = (al >= 0.0f) ? al : NEG_SLOPE * al;
        const float coef = __expf(al - m[d * H + hd]) /
                           (denom[d * H + hd] + 1e-16f);
        atomicAdd(&out[d * (long long)HC + ch], h[s * (long long)HC + ch] * coef);
    }
}

// ---------------------------------------------------------------------------
static inline unsigned int ceil_div_u(long long a, long long b) {
    return (unsigned int)((a + b - 1) / b);
}

extern "C" void kernel_launch(void* const* d_in, const int* in_sizes, int n_in,
                              void* d_out, int out_size, void* d_ws, size_t ws_size,
                              hipStream_t stream) {
    const float*     x        = (const float*)d_in[0];
    const long long* ei       = (const long long*)d_in[1];
    const float*     W1       = (const float*)d_in[2];
    const float*     att_src1 = (const float*)d_in[3];
    const float*     att_dst1 = (const float*)d_in[4];
    const float*     bias1    = (const float*)d_in[5];
    const float*     W2       = (const float*)d_in[6];
    const float*     att_src2 = (const float*)d_in[7];
    const float*     att_dst2 = (const float*)d_in[8];
    const float*     bias2    = (const float*)d_in[9];

    const int F   = 128;                       // x feature dim (== template K)
    const int H1  = 8;
    const int C1  = in_sizes[3] / H1;          // 16
    const int HC1 = in_sizes[5];               // 128 (== template M, layer 1)
    const int C2  = in_sizes[9];               // 40  (== template M, layer 2)

    const long long N  = in_sizes[0] / F;      // 50000
    const long long E  = in_sizes[1] / 2;      // 800000
    const long long EN = E + N;                // edges + self-loops

    const long long* src = ei;
    const long long* dst = ei + E;

    // ---- workspace carve (floats) ----
    float* ws = (float*)d_ws;
    size_t off = 0;
    float* h1   = ws + off; off += (size_t)N * HC1;
    float* as1  = ws + off; off += (size_t)N * H1;
    float* ad1  = ws + off; off += (size_t)N * H1;
    float* m1   = ws + off; off += (size_t)N * H1;   // uint during pass A
    float* dn1  = ws + off; off += (size_t)N * H1;
    float* out1 = ws + off; off += (size_t)N * HC1;
    float* h2   = ws + off; off += (size_t)N * C2;
    float* as2  = ws + off; off += (size_t)N;
    float* ad2  = ws + off; off += (size_t)N;
    float* m2   = ws + off; off += (size_t)N;
    float* dn2  = ws + off; off += (size_t)N;
    (void)ws_size; (void)n_in; (void)out_size;

    const unsigned int rowTiles = ceil_div_u(N, 16);
    const unsigned int eBlocks  = ceil_div_u(EN, 256);
    const unsigned int aggBlk   = ceil_div_u(EN, 8);   // 8 waves / 256-thr block

    // ================= Layer 1 =================
    // h1 = x @ W1   (M = 128 -> 8 col tiles -> 256-thread blocks)
    wmma_gemm_kernel<128, 128><<<rowTiles, 256, 0, stream>>>(x, W1, h1, (int)N);

    attn_score_kernel<<<ceil_div_u(N * H1, 256), 256, 0, stream>>>(
        h1, att_src1, att_dst1, as1, ad1, (unsigned int*)m1, dn1, N, H1, C1);

    edge_max_kernel<<<eBlocks, 256, 0, stream>>>(
        src, dst, as1, ad1, (unsigned int*)m1, E, EN, H1);

    decode_m_kernel<<<ceil_div_u(N * H1, 256), 256, 0, stream>>>(
        (unsigned int*)m1, N * H1);

    edge_sum_kernel<<<eBlocks, 256, 0, stream>>>(
        src, dst, as1, ad1, m1, dn1, E, EN, H1);

    init_bias_kernel<<<ceil_div_u(N * HC1, 256), 256, 0, stream>>>(
        out1, bias1, N, HC1);

    edge_agg_kernel<<<aggBlk, 256, 0, stream>>>(
        src, dst, as1, ad1, m1, dn1, h1, out1, E, EN, H1, C1);

    // ================= Layer 2 =================
    // h2 = out1 @ W2   (M = 40 -> 3 col tiles -> 96-thread blocks)
    wmma_gemm_kernel<128, 40><<<rowTiles, 96, 0, stream>>>(out1, W2, h2, (int)N);

    attn_score_kernel<<<ceil_div_u(N, 256), 256, 0, stream>>>(
        h2, att_src2, att_dst2, as2, ad2, (unsigned int*)m2, dn2, N, 1, C2);

    edge_max_kernel<<<eBlocks, 256, 0, stream>>>(
        src, dst, as2, ad2, (unsigned int*)m2, E, EN, 1);

    decode_m_kernel<<<ceil_div_u(N, 256), 256, 0, stream>>>(
        (unsigned int*)m2, N);

    edge_sum_kernel<<<eBlocks, 256, 0, stream>>>(
        src, dst, as2, ad2, m2, dn2, E, EN, 1);

    float* out = (float*)d_out;
    init_bias_kernel<<<ceil_div_u(N * C2, 256), 256, 0, stream>>>(
        out, bias2, N, C2);

    edge_agg_kernel<<<aggBlk, 256, 0, stream>>>(
        src, dst, as2, ad2, m2, dn2, h2, out, E, EN, 1, C2);
}